// FaceModel_83141976916300
// MI455X (gfx1250) — compile-verified
//
#include <hip/hip_runtime.h>
#include <stdint.h>

#define B_IMG 8
#define Hh 512
#define Ww 512
#define HW (Hh * Ww)            // 262144 = 2^18
#define NTOT (B_IMG * HW)       // 2097152
#define Kk 2048
#define CAP 4096
#define NBINS 1024
#define TILE 4096               // floats per TDM chunk (16 KB)
#define SCAN_BLOCKS 128
#define NCHUNK (NTOT / TILE)    // 512
#define CHUNKS_PER_BLOCK (NCHUNK / SCAN_BLOCKS)   // 4
#define NEGV -1.0e9f
#define BITS_LO 0x3F199999u     // just below bits(0.6f)

typedef unsigned int u32;
typedef unsigned int u32x4 __attribute__((ext_vector_type(4)));
typedef int i32x4 __attribute__((ext_vector_type(4)));
typedef int i32x8 __attribute__((ext_vector_type(8)));

// ---------------------------------------------------------------------------
// TDM: DMA a contiguous row of `nelem` 32-bit elements from global into LDS.
// Builds a D# (groups 0/1 per CDNA5 ISA ch.8; groups 2/3 zero => dims 2..4
// unused). Tracked on TENSORcnt.
// ---------------------------------------------------------------------------
__device__ __forceinline__ void tdm_load_row(void* lds_ptr, const void* gptr, u32 nelem) {
  u32 lds = (u32)(uintptr_t)lds_ptr;                 // low 32 bits of flat LDS addr = LDS offset
  unsigned long long ga = (unsigned long long)(uintptr_t)gptr;
  u32x4 g0;
  g0.x = 1u;                                          // count=1, user descriptor, no gather
  g0.y = lds;                                         // lds_addr
  g0.z = (u32)ga;                                     // global_addr[31:0]
  g0.w = (u32)((ga >> 32) & 0x1FFFFFFull) | (2u << 30); // global_addr[56:32] | type=2
  i32x8 g1;
  g1[0] = 0x00020000;                                 // wg_mask=0, data_size=2 (4B)
  g1[1] = (int)(nelem << 16);                         // tensor_dim0[15:0]
  g1[2] = (int)((nelem >> 16) | (1u << 16));          // tensor_dim0[31:16] | tensor_dim1=1
  g1[3] = (int)(nelem << 16);                         // tile_dim0 = nelem
  g1[4] = 1;                                          // tile_dim1=1, tile_dim2=0
  g1[5] = (int)nelem;                                 // tensor_dim0_stride[31:0]
  g1[6] = 0;
  g1[7] = 0;
  i32x4 z4 = {0, 0, 0, 0};
#if __clang_major__ >= 23
  i32x8 z8 = {0, 0, 0, 0, 0, 0, 0, 0};
  __builtin_amdgcn_tensor_load_to_lds(g0, g1, z4, z4, z8, 0);
#else
  __builtin_amdgcn_tensor_load_to_lds(g0, g1, z4, z4, 0);
#endif
}

__device__ __forceinline__ u32 score_bucket(u32 bits) {
  u32 bk = (bits - BITS_LO) >> 13;                    // [0.6,1.0) -> bins 0..819
  return (bk > NBINS - 1u) ? (NBINS - 1u) : bk;
}

// ---------------------------------------------------------------------------
// 0: zero scratch scalars + histogram
// ---------------------------------------------------------------------------
__global__ void __launch_bounds__(256) zero_kernel(u32* hist, u32* cnt, u32* thr, u32* maxc) {
  for (int t = threadIdx.x; t < NBINS; t += 256) hist[t] = 0u;
  if (threadIdx.x == 0) { *cnt = 0u; *thr = 0u; *maxc = 0u; }
}

// ---------------------------------------------------------------------------
// 1: histogram of passing-score float bits; probs streamed via TDM into LDS
//    with double buffering (DMA of chunk c+1 overlaps VALU on chunk c).
//    LDS consumed as float4 (ds_load_b128).
// ---------------------------------------------------------------------------
__global__ void __launch_bounds__(256) hist_kernel(const float* __restrict__ probs,
                                                   u32* __restrict__ hist) {
  __shared__ float buf[2][TILE];                      // 32 KB
  __shared__ u32 lh[NBINS];                           // 4 KB
  for (int t = threadIdx.x; t < NBINS; t += 256) lh[t] = 0u;
  const int base = blockIdx.x * CHUNKS_PER_BLOCK;
  if (threadIdx.x < 32) {                             // wave 0 issues TDM
    tdm_load_row(&buf[0][0], probs + (size_t)base * TILE, TILE);
  }
  __syncthreads();
  for (int c = 0; c < CHUNKS_PER_BLOCK; ++c) {
    if (threadIdx.x < 32) {
      __builtin_amdgcn_s_wait_tensorcnt(0);           // current buffer ready
    }
    __syncthreads();
    if (threadIdx.x < 32 && (c + 1) < CHUNKS_PER_BLOCK) {
      tdm_load_row(&buf[(c + 1) & 1][0], probs + (size_t)(base + c + 1) * TILE, TILE);
    }
    const float4* src4 = (const float4*)buf[c & 1];
    for (int t = threadIdx.x; t < TILE / 4; t += 256) {
      float4 v = src4[t];
      float sv[4] = {v.x, v.y, v.z, v.w};
#pragma unroll
      for (int j = 0; j < 4; ++j) {
        float s = sv[j];
        if (s >= 0.6f) atomicAdd(&lh[score_bucket(__float_as_uint(s))], 1u);
      }
    }
    __syncthreads();
  }
  for (int t = threadIdx.x; t < NBINS; t += 256) {
    u32 v = lh[t];
    if (v) atomicAdd(&hist[t], v);
  }
}

// ---------------------------------------------------------------------------
// 2: suffix-sum histogram from the top; T = highest bin with >= K items above
// ---------------------------------------------------------------------------
__global__ void __launch_bounds__(NBINS) threshold_kernel(const u32* __restrict__ hist,
                                                          u32* __restrict__ thr) {
  __shared__ u32 sh[NBINS];
  int t = threadIdx.x;
  sh[t] = hist[t];
  __syncthreads();
  for (int off = 1; off < NBINS; off <<= 1) {
    u32 v = (t + off < NBINS) ? sh[t + off] : 0u;
    __syncthreads();
    sh[t] += v;                                       // sh[t] = count of bins >= t
    __syncthreads();
  }
  if (t == 0) *thr = 0u;
  __syncthreads();
  bool ok = sh[t] >= (u32)Kk;
  bool nextOk = (t + 1 < NBINS) ? (sh[t + 1] >= (u32)Kk) : false;
  if (ok && !nextOk) *thr = (u32)t;                   // unique writer (sh monotone)
}

// ---------------------------------------------------------------------------
// 3: compact candidates in buckets >= T. Same TDM-streamed double-buffered
//    LDS pipeline as the histogram pass; float4 LDS reads.
// ---------------------------------------------------------------------------
__global__ void __launch_bounds__(256) compact_kernel(const float* __restrict__ probs,
                                                      const u32* __restrict__ thr,
                                                      u32* __restrict__ cnt,
                                                      u32* __restrict__ candBits,
                                                      u32* __restrict__ candIdx) {
  __shared__ float buf[2][TILE];                      // 32 KB
  u32 T = *thr;
  const int base = blockIdx.x * CHUNKS_PER_BLOCK;
  if (threadIdx.x < 32) {
    tdm_load_row(&buf[0][0], probs + (size_t)base * TILE, TILE);
  }
  __syncthreads();
  for (int c = 0; c < CHUNKS_PER_BLOCK; ++c) {
    if (threadIdx.x < 32) {
      __builtin_amdgcn_s_wait_tensorcnt(0);
    }
    __syncthreads();
    if (threadIdx.x < 32 && (c + 1) < CHUNKS_PER_BLOCK) {
      tdm_load_row(&buf[(c + 1) & 1][0], probs + (size_t)(base + c + 1) * TILE, TILE);
    }
    const float4* src4 = (const float4*)buf[c & 1];
    const int gbase = (base + c) * TILE;
    for (int t = threadIdx.x; t < TILE / 4; t += 256) {
      float4 v = src4[t];
      float sv[4] = {v.x, v.y, v.z, v.w};
#pragma unroll
      for (int j = 0; j < 4; ++j) {
        float s = sv[j];
        if (s >= 0.6f) {
          u32 bits = __float_as_uint(s);
          if (score_bucket(bits) >= T) {
            u32 p = atomicAdd(cnt, 1u);
            if (p < (u32)CAP) {
              candBits[p] = bits;
              candIdx[p]  = (u32)(gbase + t * 4 + j);
            }
          }
        }
      }
    }
    __syncthreads();
  }
}

// ---------------------------------------------------------------------------
// 4: single-block bitonic sort (desc, by float bits), box gen, maxc reduce,
//    per-image offsets. Writes top-K: sc, offset boxes, source indices.
// ---------------------------------------------------------------------------
__global__ void __launch_bounds__(1024) sort_kernel(const u32* __restrict__ candBits,
                                                    const u32* __restrict__ candIdx,
                                                    const u32* __restrict__ cnt,
                                                    float* __restrict__ sc,
                                                    float* __restrict__ boxes,
                                                    u32* __restrict__ idxs,
                                                    u32* __restrict__ maxcOut) {
  __shared__ u32 sb[CAP];                             // 16 KB
  __shared__ u32 si[CAP];                             // 16 KB
  __shared__ u32 smax;
  u32 n = *cnt; if (n > (u32)CAP) n = (u32)CAP;
  for (int j = threadIdx.x; j < CAP; j += 1024) {
    if ((u32)j < n) { sb[j] = candBits[j]; si[j] = candIdx[j]; }
    else            { sb[j] = 0u;          si[j] = 0u; }
  }
  if (threadIdx.x == 0) smax = 0u;
  __syncthreads();
  for (u32 k = 2; k <= (u32)CAP; k <<= 1) {
    for (u32 jj = k >> 1; jj > 0; jj >>= 1) {
      for (u32 i = threadIdx.x; i < (u32)CAP; i += 1024) {
        u32 ixj = i ^ jj;
        if (ixj > i) {
          bool up = ((i & k) == 0);
          u32 a = sb[i], b = sb[ixj];
          bool sw = up ? (a < b) : (a > b);           // descending overall
          if (sw) {
            sb[i] = b; sb[ixj] = a;
            u32 ta = si[i]; si[i] = si[ixj]; si[ixj] = ta;
          }
        }
      }
      __syncthreads();
    }
  }
  // two top-K entries per thread
  u32 myBits[2], myIdx[2];
  for (int q = 0; q < 2; ++q) {
    int kk = threadIdx.x + q * 1024;
    myBits[q] = sb[kk]; myIdx[q] = si[kk];
    if (myBits[q] != 0u) {
      u32 x = myIdx[q] & 511u, y = (myIdx[q] >> 9) & 511u;
      u32 m = 4u * x + 24u, m2 = 4u * y + 24u;        // q2 coords dominate
      if (m2 > m) m = m2;
      atomicMax(&smax, m);
    }
  }
  __syncthreads();
  float maxc = (float)smax;
  if (threadIdx.x == 0) *maxcOut = smax;
  for (int q = 0; q < 2; ++q) {
    int kk = threadIdx.x + q * 1024;
    u32 idx = myIdx[q];
    u32 x = idx & 511u, y = (idx >> 9) & 511u, bimg = idx >> 18;
    float off = (float)bimg * (maxc + 1.0f);          // batched-NMS offset
    boxes[kk * 4 + 0] = (float)(4u * x + 2u)  + off;
    boxes[kk * 4 + 1] = (float)(4u * y + 2u)  + off;
    boxes[kk * 4 + 2] = (float)(4u * x + 24u) + off;
    boxes[kk * 4 + 3] = (float)(4u * y + 24u) + off;
    sc[kk]  = (myBits[q] != 0u) ? __uint_as_float(myBits[q]) : NEGV;
    idxs[kk] = idx;
  }
}

// ---------------------------------------------------------------------------
// 5: fully parallel IoU suppression bitmatrices for both thresholds
//    (2048 rows x 64 words; one word per thread; 131072 threads)
// ---------------------------------------------------------------------------
__global__ void __launch_bounds__(256) iou_kernel(const float* __restrict__ boxes,
                                                  const float* __restrict__ sc,
                                                  u32* __restrict__ maskA,
                                                  u32* __restrict__ maskB) {
  int gid = blockIdx.x * 256 + threadIdx.x;           // 0..131071
  int r = gid >> 6;
  int w = gid & 63;
  float x1 = boxes[r * 4 + 0], y1 = boxes[r * 4 + 1];
  float x2 = boxes[r * 4 + 2], y2 = boxes[r * 4 + 3];
  bool vr = sc[r] > -5.0e8f;
  float ar = (x2 - x1 + 1.0f) * (y2 - y1 + 1.0f);
  u32 mA = 0u, mB = 0u;
  int cbase = w * 32;
  for (int t = 0; t < 32; ++t) {
    int c = cbase + t;
    float cx1 = boxes[c * 4 + 0], cy1 = boxes[c * 4 + 1];
    float cx2 = boxes[c * 4 + 2], cy2 = boxes[c * 4 + 3];
    float xx1 = fmaxf(x1, cx1), yy1 = fmaxf(y1, cy1);
    float xx2 = fminf(x2, cx2), yy2 = fminf(y2, cy2);
    float iw = fmaxf(0.0f, xx2 - xx1 + 1.0f);
    float ih = fmaxf(0.0f, yy2 - yy1 + 1.0f);
    float inter = iw * ih;
    float ac = (cx2 - cx1 + 1.0f) * (cy2 - cy1 + 1.0f);
    float iou = inter / (ar + ac - inter);
    if (vr && iou > 0.5f) mA |= (1u << t);
    if (vr && iou > 0.7f) mB |= (1u << t);
  }
  maskA[(size_t)r * 64 + w] = mA;
  maskB[(size_t)r * 64 + w] = mB;
}

// ---------------------------------------------------------------------------
// 6: serial (but bitmask-cheap) greedy NMS x2, then bbreg + rerec + output
// ---------------------------------------------------------------------------
__global__ void __launch_bounds__(256) final_kernel(const float* __restrict__ regp,
                                                    const float* __restrict__ sc,
                                                    const u32* __restrict__ idxs,
                                                    const u32* __restrict__ maskA,
                                                    const u32* __restrict__ maskB,
                                                    float* __restrict__ out) {
  __shared__ float ssc[Kk];                           // 8 KB
  __shared__ u32 removed[Kk / 32];
  __shared__ u32 keep1[Kk / 32];
  __shared__ u32 keep2[Kk / 32];
  int tid = threadIdx.x;
  for (int k = tid; k < Kk; k += 256) ssc[k] = sc[k];
  for (int t = tid; t < Kk / 32; t += 256) { removed[t] = 0u; keep1[t] = 0u; keep2[t] = 0u; }
  __syncthreads();
  // pass 1: IoU > 0.5
  for (int k = 0; k < Kk; ++k) {
    bool rem  = (removed[k >> 5] >> (k & 31)) & 1u;
    bool kept = (!rem) && (ssc[k] > -5.0e8f);
    __syncthreads();                                  // snapshot reads before writes
    if (kept) {
      if (tid < Kk / 32) removed[tid] |= maskA[(size_t)k * 64 + tid];
      if (tid == 0) keep1[k >> 5] |= (1u << (k & 31));
    }
    __syncthreads();
  }
  for (int t = tid; t < Kk / 32; t += 256) removed[t] = 0u;
  __syncthreads();
  // pass 2: IoU > 0.7 on keep1 survivors
  for (int k = 0; k < Kk; ++k) {
    bool rem  = (removed[k >> 5] >> (k & 31)) & 1u;
    bool k1   = (keep1[k >> 5] >> (k & 31)) & 1u;
    bool kept = (!rem) && k1;
    __syncthreads();
    if (kept) {
      if (tid < Kk / 32) removed[tid] |= maskB[(size_t)k * 64 + tid];
      if (tid == 0) keep2[k >> 5] |= (1u << (k & 31));
    }
    __syncthreads();
  }
  __syncthreads();
  // bbreg + rerec + gated output
  for (int k = tid; k < Kk; k += 256) {
    u32 idx = idxs[k];
    u32 x = idx & 511u, y = (idx >> 9) & 511u, bimg = idx >> 18;
    float fx1 = (float)(4u * x + 2u),  fy1 = (float)(4u * y + 2u);
    float fx2 = (float)(4u * x + 24u), fy2 = (float)(4u * y + 24u);
    size_t baseo = ((size_t)bimg * 4u) * (size_t)HW + (size_t)y * Ww + x;
    float r0 = regp[baseo];
    float r1 = regp[baseo + (size_t)HW];
    float r2 = regp[baseo + 2u * (size_t)HW];
    float r3 = regp[baseo + 3u * (size_t)HW];
    float w_ = fx2 - fx1, h_ = fy2 - fy1;             // constant 22
    float q1 = fx1 + r0 * w_;
    float q2 = fy1 + r1 * h_;
    float q3 = fx2 + r2 * w_;
    float q4 = fy2 + r3 * h_;
    float wwv = q3 - q1, hhv = q4 - q2;
    float l = fmaxf(wwv, hhv);
    float x0 = q1 + wwv * 0.5f - l * 0.5f;
    float y0 = q2 + hhv * 0.5f - l * 0.5f;
    float kf = ((keep2[k >> 5] >> (k & 31)) & 1u) ? 1.0f : 0.0f;
    out[(size_t)k * 5 + 0] = x0 * kf;
    out[(size_t)k * 5 + 1] = y0 * kf;
    out[(size_t)k * 5 + 2] = (x0 + l) * kf;
    out[(size_t)k * 5 + 3] = (y0 + l) * kf;
    out[(size_t)k * 5 + 4] = ssc[k] * kf;
  }
}

extern "C" void kernel_launch(void* const* d_in, const int* in_sizes, int n_in,
                              void* d_out, int out_size, void* d_ws, size_t ws_size,
                              hipStream_t stream) {
  const float* regp  = (const float*)d_in[0];         // [8,4,512,512]
  const float* probs = (const float*)d_in[1];         // [8,512,512]
  float* out = (float*)d_out;                         // [2048,5]

  u32* ws       = (u32*)d_ws;
  u32* hist     = ws;                                 // 1024
  u32* cnt      = hist + NBINS;                       // 1
  u32* thr      = cnt + 1;                            // 1
  u32* maxc     = thr + 1;                            // 1
  u32* candBits = hist + NBINS + 16;                  // 4096
  u32* candIdx  = candBits + CAP;                     // 4096
  float* scArr  = (float*)(candIdx + CAP);            // 2048
  float* boxes  = scArr + Kk;                         // 2048*4
  u32* idxs     = (u32*)(boxes + (size_t)Kk * 4);     // 2048
  u32* maskA    = idxs + Kk;                          // 2048*64
  u32* maskB    = maskA + (size_t)Kk * 64;            // 2048*64
  (void)in_sizes; (void)n_in; (void)out_size; (void)ws_size;

  zero_kernel<<<1, 256, 0, stream>>>(hist, cnt, thr, maxc);
  hist_kernel<<<SCAN_BLOCKS, 256, 0, stream>>>(probs, hist);
  threshold_kernel<<<1, NBINS, 0, stream>>>(hist, thr);
  compact_kernel<<<SCAN_BLOCKS, 256, 0, stream>>>(probs, thr, cnt, candBits, candIdx);
  sort_kernel<<<1, 1024, 0, stream>>>(candBits, candIdx, cnt, scArr, boxes, idxs, maxc);
  iou_kernel<<<512, 256, 0, stream>>>(boxes, scArr, maskA, maskB);
  final_kernel<<<1, 256, 0, stream>>>(regp, scArr, idxs, maskA, maskB, out);
}